// NLL_84559316124181
// MI455X (gfx1250) — compile-verified
//
#include <hip/hip_runtime.h>
#include <math.h>

// Problem constants (from reference)
#define NPTS 4096
#define Q0C  1000
#define Q1C  500
#define QTOT 1500
#define BS   64             // Cholesky block size
#define PP   1536           // QTOT padded to multiple of BS (identity padding)
#define NB   (PP / BS)      // 24 blocks

typedef float v2f __attribute__((ext_vector_type(2)));
typedef float v8f __attribute__((ext_vector_type(8)));

// ---------------------------------------------------------------------------
// 1) Initialize M = D^{-1} (diag), identity in padding rows; zero u, scalars.
// ---------------------------------------------------------------------------
__global__ void init_M(float* __restrict__ M, float* __restrict__ u,
                       float* __restrict__ scal, const float* __restrict__ sig2bs) {
    size_t idx = (size_t)blockIdx.x * blockDim.x + threadIdx.x;
    size_t total = (size_t)PP * PP;
    if (idx < total) {
        size_t row = idx / PP, col = idx % PP;
        float v = 0.0f;
        if (row == col) {
            if (row < Q0C)            v = 1.0f / sig2bs[0];
            else if (row < QTOT)      v = 1.0f / sig2bs[1];
            else                      v = 1.0f;   // padding -> L_ii = 1, log adds 0
        }
        M[idx] = v;
    }
    if (idx < PP) u[idx] = 0.0f;
    if (idx < 2)  scal[idx] = 0.0f;   // [0]=r^T r, [1]=sum log L_ii
}

// ---------------------------------------------------------------------------
// 2) Scatter:  M += sig2e^{-1} * Z^T Z  (lower triangle only),  u = Z^T r,
//              scal[0] = r^T r
// ---------------------------------------------------------------------------
__global__ void build_scatter(const float* __restrict__ yt, const float* __restrict__ yp,
                              const int* __restrict__ z0, const int* __restrict__ z1,
                              const float* __restrict__ sig2e_p,
                              float* __restrict__ M, float* __restrict__ u,
                              float* __restrict__ scal, int n) {
    int idx = blockIdx.x * blockDim.x + threadIdx.x;
    if (idx >= n) return;
    float inv_e = 1.0f / sig2e_p[0];
    float r = yt[idx] - yp[idx];
    int a = z0[idx];
    int b = Q0C + z1[idx];            // b > a always (cross entry lands in lower tri)
    atomicAdd(&M[(size_t)a * PP + a], inv_e);
    atomicAdd(&M[(size_t)b * PP + b], inv_e);
    atomicAdd(&M[(size_t)b * PP + a], inv_e);
    atomicAdd(&u[a], r);
    atomicAdd(&u[b], r);
    atomicAdd(&scal[0], r * r);
}

// ---------------------------------------------------------------------------
// 3a) Cholesky of 64x64 diagonal block (64 threads, row per thread);
//     accumulates sum(log L_ii).  Only lower triangle is read/written.
// ---------------------------------------------------------------------------
__global__ void chol_diag(float* __restrict__ M, float* __restrict__ logacc, int k) {
    __shared__ float s[BS][BS + 1];           // +1 pad: conflict-free columns
    float* A = M + (size_t)(k * BS) * PP + k * BS;
    int t = threadIdx.x;                      // 0..63, row index
    for (int j = 0; j < BS; ++j) s[t][j] = A[(size_t)t * PP + j];
    __syncthreads();
    for (int j = 0; j < BS; ++j) {
        if (t == j) s[j][j] = sqrtf(s[j][j]);
        __syncthreads();
        float d = s[j][j];
        if (t > j) s[t][j] /= d;
        __syncthreads();
        if (t > j)
            for (int m = j + 1; m <= t; ++m) s[t][m] -= s[t][j] * s[m][j];
        __syncthreads();
    }
    for (int j = 0; j <= t; ++j) A[(size_t)t * PP + j] = s[t][j];
    atomicAdd(logacc, logf(s[t][t]));
}

// ---------------------------------------------------------------------------
// 3b) Panel TRSM:  A[i][k] <- A[i][k] * L_kk^{-T}   (64 threads per block row;
//     per-row state kept in LDS to avoid scratch-spilled indexed arrays)
// ---------------------------------------------------------------------------
__global__ void panel_trsm(float* __restrict__ M, int k) {
    int i = k + 1 + blockIdx.x;
    __shared__ float Lkk[BS][BS + 1];
    __shared__ float X[BS][BS + 1];
    int t = threadIdx.x;                      // 0..63, row within panel block
    const float* Ld = M + (size_t)(k * BS) * PP + k * BS;
    for (int e = t; e < BS * BS; e += BS)
        Lkk[e >> 6][e & 63] = Ld[(size_t)(e >> 6) * PP + (e & 63)];
    float* A = M + (size_t)(i * BS) * PP + k * BS;
    for (int j = 0; j < BS; ++j) X[t][j] = A[(size_t)t * PP + j];
    __syncthreads();
    for (int j = 0; j < BS; ++j) {            // X L^T = A  (forward over columns)
        float s = X[t][j];
        for (int m = 0; m < j; ++m) s -= X[t][m] * Lkk[j][m];
        X[t][j] = s / Lkk[j][j];
    }
    for (int j = 0; j < BS; ++j) A[(size_t)t * PP + j] = X[t][j];
}

// ---------------------------------------------------------------------------
// 3c) SYRK trailing update with WMMA f32:  C(i,j) -= P_i * P_j^T.
//     64x64 tile per workgroup (4 waves); panels staged in LDS; each wave
//     owns a 16x64 strip -> 16 K-steps x 4 subtiles of
//     v_wmma_f32_16x16x4_f32 with independent accumulators (straight-line:
//     no per-subtile guards -> no EXEC juggling, no accumulator copies).
//     Diagonal 64-blocks also compute their strictly-upper subtiles; those
//     values are defined (init_M wrote the full matrix) and never read.
//     f32 WMMA NEG bits only allow C-negate (ISA 7.12) -> negate A at load.
// ---------------------------------------------------------------------------
__global__ void trailing_update(float* __restrict__ M, int k) {
    int bj = blockIdx.x, bi = blockIdx.y;
    if (bj > bi) return;                      // blockIdx-based: scalar branch
    int i64 = (k + 1 + bi) * BS;
    int j64 = (k + 1 + bj) * BS;
    int k64 = k * BS;

    __shared__ float LPi[BS][BS + 1];         // stride 65: column reads hit
    __shared__ float LPj[BS][BS + 1];         // 64 distinct banks
    int tid = threadIdx.x;                    // 0..127 (4 waves)
    const float* Pi = M + (size_t)i64 * PP + k64;
    const float* Pj = M + (size_t)j64 * PP + k64;
    for (int e = tid; e < BS * BS; e += 128) {
        int r = e >> 6, c = e & 63;
        LPi[r][c] = Pi[(size_t)r * PP + c];
        LPj[r][c] = Pj[(size_t)r * PP + c];
    }
    __syncthreads();

    int wave = tid >> 5, lane = tid & 31;
    int col  = lane & 15;                     // N for B/C/D, M-row for A
    int half = lane >> 4;
    int sr   = wave;                          // subtile row 0..3 (16 rows each)

    float* Crow = M + (size_t)(i64 + sr * 16) * PP + j64;
    v8f c0, c1, c2, c3;
#pragma unroll
    for (int r = 0; r < 8; ++r) {
        size_t ro = (size_t)(r + 8 * half) * PP + col;
        c0[r] = Crow[ro];
        c1[r] = Crow[ro + 16];
        c2[r] = Crow[ro + 32];
        c3[r] = Crow[ro + 48];
    }

#pragma unroll
    for (int kk = 0; kk < 16; ++kk) {
        v2f a, b0, b1, b2, b3;
#pragma unroll
        for (int v = 0; v < 2; ++v) {         // K = kk*4 + v + 2*half (ISA layout)
            int kidx = kk * 4 + v + 2 * half;
            a[v]  = -LPi[sr * 16 + col][kidx];    // negated A => C - Pi*Pj^T
            b0[v] = LPj[col][kidx];               // B = Pj^T, subtile columns
            b1[v] = LPj[16 + col][kidx];
            b2[v] = LPj[32 + col][kidx];
            b3[v] = LPj[48 + col][kidx];
        }
        c0 = __builtin_amdgcn_wmma_f32_16x16x4_f32(false, a, false, b0, (short)0, c0, false, false);
        c1 = __builtin_amdgcn_wmma_f32_16x16x4_f32(false, a, false, b1, (short)0, c1, false, false);
        c2 = __builtin_amdgcn_wmma_f32_16x16x4_f32(false, a, false, b2, (short)0, c2, false, false);
        c3 = __builtin_amdgcn_wmma_f32_16x16x4_f32(false, a, false, b3, (short)0, c3, false, false);
    }

#pragma unroll
    for (int r = 0; r < 8; ++r) {
        size_t ro = (size_t)(r + 8 * half) * PP + col;
        Crow[ro]      = c0[r];
        Crow[ro + 16] = c1[r];
        Crow[ro + 32] = c2[r];
        Crow[ro + 48] = c3[r];
    }
}

// ---------------------------------------------------------------------------
// 4) Triangular solves (L y = u, L^T x = y), quad = u.x, assemble NLL.
//    Single wave; lane-parallel dot products with wave reductions.
// ---------------------------------------------------------------------------
__global__ void solve_finalize(const float* __restrict__ M, const float* __restrict__ u,
                               const float* __restrict__ scal,
                               const float* __restrict__ sig2e_p,
                               const float* __restrict__ sig2bs,
                               float* __restrict__ out) {
    __shared__ float y[PP];
    int lane = threadIdx.x;                    // 32 lanes (wave32)
    // forward: L y = u
    for (int i = 0; i < PP; ++i) {
        const float* Li = M + (size_t)i * PP;
        float s = 0.0f;
        for (int j = lane; j < i; j += 32) s += Li[j] * y[j];
        for (int off = 16; off; off >>= 1) s += __shfl_down(s, off);
        if (lane == 0) y[i] = (u[i] - s) / Li[i];
        __syncthreads();
    }
    // backward: L^T x = y (in place)
    for (int i = PP - 1; i >= 0; --i) {
        float s = 0.0f;
        for (int j = i + 1 + lane; j < PP; j += 32) s += M[(size_t)j * PP + i] * y[j];
        for (int off = 16; off; off >>= 1) s += __shfl_down(s, off);
        if (lane == 0) y[i] = (y[i] - s) / M[(size_t)i * PP + i];
        __syncthreads();
    }
    // quad = u^T M^{-1} u
    float q = 0.0f;
    for (int j = lane; j < PP; j += 32) q += u[j] * y[j];
    for (int off = 16; off; off >>= 1) q += __shfl_down(q, off);

    if (lane == 0) {
        float sig2e = sig2e_p[0];
        float inv_e = 1.0f / sig2e;
        float rTr = scal[0];
        float logacc = scal[1];                // sum log L_ii
        // logdet V = N log s2e + Q0 log s2b0 + Q1 log s2b1 + 2*sum log L_ii
        float logdet = (float)NPTS * logf(sig2e)
                     + (float)Q0C  * logf(sig2bs[0])
                     + (float)Q1C  * logf(sig2bs[1])
                     + 2.0f * logacc;
        // r^T V^{-1} r = inv_e * rTr - inv_e^2 * quad   (Woodbury)
        float quad = inv_e * rTr - inv_e * inv_e * q;
        const float LOG2PI = 1.8378770664093453f;
        out[0] = 0.5f * (float)NPTS * LOG2PI + 0.5f * logdet + 0.5f * quad;
    }
}

// ---------------------------------------------------------------------------
extern "C" void kernel_launch(void* const* d_in, const int* in_sizes, int n_in,
                              void* d_out, int out_size, void* d_ws, size_t ws_size,
                              hipStream_t stream) {
    const float* y_true = (const float*)d_in[0];
    const float* y_pred = (const float*)d_in[1];
    const float* sig2e  = (const float*)d_in[2];
    const float* sig2bs = (const float*)d_in[3];
    const int*   z0     = (const int*)d_in[4];
    const int*   z1     = (const int*)d_in[5];
    float* out = (float*)d_out;

    // workspace: M (PP*PP f32, ~9.4 MB) | u (PP f32) | scal (2 f32)
    float* M    = (float*)d_ws;
    float* u    = M + (size_t)PP * PP;
    float* scal = u + PP;
    int n = in_sizes[0];

    {
        size_t total = (size_t)PP * PP;
        init_M<<<(unsigned)((total + 255) / 256), 256, 0, stream>>>(M, u, scal, sig2bs);
    }
    build_scatter<<<(n + 255) / 256, 256, 0, stream>>>(y_true, y_pred, z0, z1,
                                                       sig2e, M, u, scal, n);
    for (int k = 0; k < NB; ++k) {
        chol_diag<<<1, BS, 0, stream>>>(M, scal + 1, k);
        int T = NB - 1 - k;
        if (T > 0) {
            panel_trsm<<<T, BS, 0, stream>>>(M, k);
            dim3 grid(T, T);
            trailing_update<<<grid, 128, 0, stream>>>(M, k);
        }
    }
    solve_finalize<<<1, 32, 0, stream>>>(M, u, scal, sig2e, sig2bs, out);
}